// ControlFlowScan2Carried_22445499089115
// MI455X (gfx1250) — compile-verified
//
#include <hip/hip_runtime.h>

typedef float v2f __attribute__((ext_vector_type(2)));
typedef float v8f __attribute__((ext_vector_type(8)));

#define T_LEN 8192
#define D_DIM 4096
#define NCHUNK 64
#define ROWS_PER_CHUNK (T_LEN / NCHUNK)        // 128
#define TILES_PER_CHUNK (ROWS_PER_CHUNK / 16)  // 8

// ---------------------------------------------------------------------------
// Pass 1: per-(chunk, column) aggregates: sum(x) and prod(2x) over the chunk.
// Coalesced: consecutive threads -> consecutive columns.
// ---------------------------------------------------------------------------
__global__ void __launch_bounds__(256) scan_pass1(const float* __restrict__ x,
                                                  float* __restrict__ aggS,
                                                  float* __restrict__ aggP) {
    const int d = blockIdx.x * 256 + threadIdx.x;
    const int c = blockIdx.y;
    const float* p = x + (size_t)c * ROWS_PER_CHUNK * D_DIM + d;
    float s = 0.0f, pr = 1.0f;
#pragma unroll 8
    for (int r = 0; r < ROWS_PER_CHUNK; ++r) {
        float v = p[(size_t)r * D_DIM];
        s += v;
        pr *= 2.0f * v;
    }
    aggS[c * D_DIM + d] = s;
    aggP[c * D_DIM + d] = pr;
}

// ---------------------------------------------------------------------------
// Pass 2: exclusive prefix over the 64 chunk aggregates, in place.
// One thread per column (4096 threads total) -- tiny kernel.
// ---------------------------------------------------------------------------
__global__ void __launch_bounds__(256) scan_pass2(float* __restrict__ aggS,
                                                  float* __restrict__ aggP) {
    const int d = blockIdx.x * 256 + threadIdx.x;
    float es = 0.0f, ep = 1.0f;
    for (int c = 0; c < NCHUNK; ++c) {
        float s = aggS[c * D_DIM + d];
        float p = aggP[c * D_DIM + d];
        aggS[c * D_DIM + d] = es;
        aggP[c * D_DIM + d] = ep;
        es += s;
        ep *= p;
    }
}

// ---------------------------------------------------------------------------
// Pass 3: apply. One wave owns a (chunk, 16-column group). For each 16x16
// tile (16 rows of t, 16 cols of d):
//   * stage tile into LDS (coalesced)
//   * cumsum via 4x V_WMMA_F32_16X16X4_F32 with triangular A built in regs
//   * cumprod via split-halves per column + wave32 shuffle join
// ---------------------------------------------------------------------------
__global__ void __launch_bounds__(128) scan_pass3(const float* __restrict__ x,
                                                  const float* __restrict__ preS,
                                                  const float* __restrict__ preP,
                                                  float* __restrict__ carry1,
                                                  float* __restrict__ carry2,
                                                  float* __restrict__ out1,
                                                  float* __restrict__ out2) {
    __shared__ float tile[4][16][17];  // 17: pad to avoid bank conflicts

    const int wid  = threadIdx.x >> 5;
    const int lane = threadIdx.x & 31;
    const int half = lane >> 4;   // 0: rows 0-7 / K 0-1, 1: rows 8-15 / K 2-3
    const int n    = lane & 15;   // column within group; also A-matrix row M

    const int wlin   = blockIdx.x * 4 + wid;
    const int chunk  = wlin / (D_DIM / 16);
    const int colgrp = wlin % (D_DIM / 16);
    const int d0     = colgrp * 16;

    float carryS = preS[chunk * D_DIM + d0 + n];
    float carryP = preP[chunk * D_DIM + d0 + n];

    for (int tl = 0; tl < TILES_PER_CHUNK; ++tl) {
        const int t0 = chunk * ROWS_PER_CHUNK + tl * 16;

        // ---- stage 16x16 tile into LDS (two 64B segments per iteration) ----
#pragma unroll
        for (int i = 0; i < 8; ++i) {
            const int row = 2 * i + half;
            tile[wid][row][n] = x[(size_t)(t0 + row) * D_DIM + d0 + n];
        }

        // ---- cumsum tile: Y = L(16x16, lower-tri ones) @ X(16x16) ----------
        // A layout (16x4 f32): lanes 0-15 -> K=0,1 ; lanes 16-31 -> K=2,3
        // B layout (4x16 f32): lanes 0-15 -> K=0,1 ; lanes 16-31 -> K=2,3
        v8f acc = {};
#pragma unroll
        for (int j = 0; j < 4; ++j) {
            const int k0 = 4 * j + 2 * half;
            v2f a, b;
            a.x = (n >= k0)     ? 1.0f : 0.0f;   // L[n][k0]
            a.y = (n >= k0 + 1) ? 1.0f : 0.0f;   // L[n][k0+1]
            b.x = tile[wid][k0][n];              // X[k0][n]
            b.y = tile[wid][k0 + 1][n];          // X[k0+1][n]
            acc = __builtin_amdgcn_wmma_f32_16x16x4_f32(
                false, a, false, b, (short)0, acc, false, false);
        }

        // acc[r] holds Y[8*half + r][n]; add running column carry, store out1.
#pragma unroll
        for (int r = 0; r < 8; ++r) {
            const float v = acc[r] + carryS;
            const int row = 8 * half + r;
            out1[(size_t)(t0 + row) * D_DIM + d0 + n] = v;
            acc[r] = v;
        }
        // new carry = inclusive sum at row 15 (lives in lane 16+n, VGPR 7)
        carryS = __shfl(acc[7], 16 + n, 32);

        // ---- cumprod: each half-lane scans 8 rows of its column ------------
        float q[8];
        float run = 1.0f;
        const int rb = 8 * half;
#pragma unroll
        for (int i = 0; i < 8; ++i) {
            run *= 2.0f * tile[wid][rb + i][n];
            q[i] = run;
        }
        // half-1 lanes need the half-0 total for their column (lane n)
        const float lowTot = __shfl(run, n, 32);
        const float scale  = half ? (carryP * lowTot) : carryP;
        float top = 0.0f;
#pragma unroll
        for (int i = 0; i < 8; ++i) {
            const float v = scale * q[i];
            out2[(size_t)(t0 + rb + i) * D_DIM + d0 + n] = v;
            top = v;
        }
        // new carry = inclusive product at row 15 (lane 16+n)
        carryP = __shfl(top, 16 + n, 32);
    }

    if (chunk == NCHUNK - 1 && half == 0) {
        carry1[d0 + n] = carryS;
        carry2[d0 + n] = carryP;
    }
}

// ---------------------------------------------------------------------------
extern "C" void kernel_launch(void* const* d_in, const int* in_sizes, int n_in,
                              void* d_out, int out_size, void* d_ws, size_t ws_size,
                              hipStream_t stream) {
    const float* x = (const float*)d_in[0];
    float* out = (float*)d_out;

    // d_out layout: carry1[D] | carry2[D] | out1[T*D] | out2[T*D]
    float* carry1 = out;
    float* carry2 = out + D_DIM;
    float* out1   = out + 2 * (size_t)D_DIM;
    float* out2   = out + 2 * (size_t)D_DIM + (size_t)T_LEN * D_DIM;

    // workspace: chunk aggregates / exclusive prefixes (2 MiB)
    float* aggS = (float*)d_ws;
    float* aggP = aggS + (size_t)NCHUNK * D_DIM;

    scan_pass1<<<dim3(D_DIM / 256, NCHUNK), dim3(256), 0, stream>>>(x, aggS, aggP);
    scan_pass2<<<dim3(D_DIM / 256), dim3(256), 0, stream>>>(aggS, aggP);

    const int nwaves = NCHUNK * (D_DIM / 16);  // 16384 waves
    scan_pass3<<<dim3(nwaves / 4), dim3(128), 0, stream>>>(
        x, aggS, aggP, carry1, carry2, out1, out2);
}